// Block_16621523436203
// MI455X (gfx1250) — compile-verified
//
#include <hip/hip_runtime.h>
#include <math.h>

// ---------------- problem constants ----------------
#define BB 8
#define LL 1024
#define DM 512           // d_model
#define DI 1024          // d_inner
#define D2C 512          // per-branch channels
#define DTR 32           // dt_rank
#define NST 16           // d_state
#define MROWS (BB * LL)  // 8192
#define HID 2048
#define SEG 16           // scan segments
#define SLEN (LL / SEG)  // 64 steps per segment

typedef float v2f __attribute__((ext_vector_type(2)));
typedef float v8f __attribute__((ext_vector_type(8)));

// ---------------- LayerNorm (one block per row of 512) ----------------
__global__ __launch_bounds__(128) void ln_kernel(const float* __restrict__ X,
                                                 const float* __restrict__ w,
                                                 const float* __restrict__ b,
                                                 float* __restrict__ Y) {
  const int row = blockIdx.x;
  const int tid = threadIdx.x;
  const float4 v = ((const float4*)(X + (size_t)row * DM))[tid];
  float s = v.x + v.y + v.z + v.w;
  float sq = v.x * v.x + v.y * v.y + v.z * v.z + v.w * v.w;
  __shared__ float sbuf[128];
  __shared__ float qbuf[128];
  sbuf[tid] = s; qbuf[tid] = sq;
  __syncthreads();
  for (int off = 64; off > 0; off >>= 1) {
    if (tid < off) { sbuf[tid] += sbuf[tid + off]; qbuf[tid] += qbuf[tid + off]; }
    __syncthreads();
  }
  const float mean = sbuf[0] * (1.0f / DM);
  const float var  = qbuf[0] * (1.0f / DM) - mean * mean;
  const float inv  = rsqrtf(var + 1e-5f);
  const float4 wv = ((const float4*)w)[tid];
  const float4 bv = ((const float4*)b)[tid];
  float4 o;
  o.x = (v.x - mean) * inv * wv.x + bv.x;
  o.y = (v.y - mean) * inv * wv.y + bv.y;
  o.z = (v.z - mean) * inv * wv.z + bv.z;
  o.w = (v.w - mean) * inv * wv.w + bv.w;
  ((float4*)(Y + (size_t)row * DM))[tid] = o;
}

// ---------------- WMMA fp32 GEMM: C[M,N] = A[M,K] * W[N,K]^T (+ epilogue) ----
// Block = 8 waves = 128(M) x 64(N); wave = 16(M) x 64(N) via 4 WMMA tiles.
// Weight tile (64 x KC) staged to LDS with global_load_async_to_lds_b128,
// double-buffered; one s_wait_asynccnt + barrier per K-chunk.
// EPI: 0=none, 1=bias+softplus, 2=+resid, 3=bias+gelu(exact), 4=bias+resid
template <int EPI, int KC>
__global__ __launch_bounds__(256) void gemm_wmma_f32(
    const float* __restrict__ A, int lda,
    const float* __restrict__ W, int ldw, int K,
    const float* __restrict__ bias, const float* __restrict__ resid,
    float* __restrict__ C, int N) {
  constexpr int STR = KC + 4;  // padded LDS row stride (floats) -> bank-spread
  __shared__ __align__(16) float wt[2][64 * STR];

  const int tid  = threadIdx.x;
  const int lane = tid & 31;
  const int wv   = tid >> 5;
  const int hi   = lane >> 4;       // half-wave K-pair select
  const int l16  = lane & 15;
  const int mrow = blockIdx.y * 128 + wv * 16 + l16;   // A row for this lane
  const int n0   = blockIdx.x * 64;
  const float* Arow = A + (size_t)mrow * lda + 2 * hi;

  v8f acc[4] = {};

  // stage one 64 x KC weight chunk into LDS buffer `buf` (async DMA)
  auto stage = [&](int kchunk, int buf) {
    constexpr int SPR = KC / 4;              // 16B segments per row
    constexpr int PT  = (64 * SPR) / 256;    // segments per thread
#pragma unroll
    for (int i = 0; i < PT; ++i) {
      const int s   = tid * PT + i;
      const int row = s / SPR;
      const int c4  = s % SPR;
      const unsigned lo = (unsigned)(size_t)&wt[buf][row * STR + c4 * 4];
      const unsigned long long ga = (unsigned long long)(size_t)(
          W + (size_t)(n0 + row) * ldw + kchunk + c4 * 4);
      asm volatile("global_load_async_to_lds_b128 %0, %1, off"
                   :: "v"(lo), "v"(ga) : "memory");
    }
  };

  const int nchunks = K / KC;
  stage(0, 0);
  asm volatile("s_wait_asynccnt 0x0" ::: "memory");
  __syncthreads();

  for (int c = 0; c < nchunks; ++c) {
    const int buf = c & 1;
    if (c + 1 < nchunks) stage((c + 1) * KC, buf ^ 1);

    // burst-load this chunk's A fragments into registers
    v2f areg[KC / 4];
#pragma unroll
    for (int i = 0; i < KC / 4; ++i)
      areg[i] = *(const v2f*)(Arow + c * KC + i * 4);

#pragma unroll
    for (int i = 0; i < KC / 4; ++i) {
#pragma unroll
      for (int t = 0; t < 4; ++t) {
        const v2f bfrag =
            *(const v2f*)&wt[buf][(t * 16 + l16) * STR + i * 4 + 2 * hi];
        acc[t] = __builtin_amdgcn_wmma_f32_16x16x4_f32(
            /*neg_a=*/false, areg[i], /*neg_b=*/false, bfrag,
            /*c_mod=*/(short)0, acc[t], /*reuse_a=*/false, /*reuse_b=*/false);
      }
    }
    if (c + 1 < nchunks) {
      asm volatile("s_wait_asynccnt 0x0" ::: "memory");
      __syncthreads();  // chunk c+1 landed; all waves done reading buf
    }
  }

  // C/D layout: VGPR r -> M = wavebase + r + 8*hi, N = tilebase + (lane&15)
#pragma unroll
  for (int r = 0; r < 8; ++r) {
    const int ms = blockIdx.y * 128 + wv * 16 + r + 8 * hi;
    const size_t rowoff = (size_t)ms * N;
#pragma unroll
    for (int t = 0; t < 4; ++t) {
      const int ns = n0 + t * 16 + l16;
      float v = acc[t][r];
      if (EPI == 1) {                       // bias + softplus
        v += bias[ns];
        v = (v > 20.0f) ? v : log1pf(expf(v));
      } else if (EPI == 2) {                // + residual
        v += resid[rowoff + ns];
      } else if (EPI == 3) {                // bias + exact GELU
        v += bias[ns];
        v = 0.5f * v * (1.0f + erff(v * 0.70710678118654752f));
      } else if (EPI == 4) {                // bias + residual
        v += bias[ns] + resid[rowoff + ns];
      }
      C[rowoff + ns] = v;
    }
  }
}

// ---------------- depthwise conv (K=4, pad 1/2) + SiLU on both branches ----
__global__ __launch_bounds__(256) void conv_silu_kernel(
    const float* __restrict__ xz, const float* __restrict__ cwx,
    const float* __restrict__ cwz, float* __restrict__ xh,
    float* __restrict__ yz) {
  const int idx = blockIdx.x * 256 + threadIdx.x;  // m*512 + d
  const int m = idx >> 9;
  const int d = idx & 511;
  const int b = m >> 10;
  const int l = m & 1023;
  float ax = 0.0f, az = 0.0f;
#pragma unroll
  for (int k = 0; k < 4; ++k) {
    const int ll = l + k - 1;
    if (ll >= 0 && ll < LL) {
      const size_t row = (size_t)(b * LL + ll) * DI;
      ax += cwx[d * 4 + k] * xz[row + d];
      az += cwz[d * 4 + k] * xz[row + D2C + d];
    }
  }
  ax = ax / (1.0f + expf(-ax));   // SiLU
  az = az / (1.0f + expf(-az));
  xh[(size_t)m * D2C + d] = ax;
  yz[(size_t)m * DI + D2C + d] = az;
}

// ---------------- segmented selective scan ----------------
// Buffers P,q,hstart laid out (seg, b, n, d) -> all accesses d-coalesced.
// Pass 1: per-segment P[n] = prod dA, q[n] = local scan end state.
__global__ __launch_bounds__(256) void scan_seg_kernel(
    const float* __restrict__ delta, const float* __restrict__ u,
    const float* __restrict__ xdbl, const float* __restrict__ A_log,
    float* __restrict__ Pbuf, float* __restrict__ qbuf) {
  const int t = blockIdx.x * 256 + threadIdx.x;  // 65536
  const int d = t & 511;
  const int b = (t >> 9) & 7;
  const int s = t >> 12;
  float Areg[NST], P[NST], q[NST];
#pragma unroll
  for (int n = 0; n < NST; ++n) {
    Areg[n] = -expf(A_log[d * NST + n]);
    P[n] = 1.0f; q[n] = 0.0f;
  }
  for (int l = s * SLEN; l < s * SLEN + SLEN; ++l) {
    const size_t row = (size_t)(b * LL + l);
    const float dl = delta[row * D2C + d];
    const float ul = u[row * D2C + d];
    const float du = dl * ul;
    const float* xd = xdbl + row * 64;
#pragma unroll
    for (int n = 0; n < NST; ++n) {
      const float dA = expf(dl * Areg[n]);
      P[n] *= dA;
      q[n] = dA * q[n] + du * xd[DTR + n];
    }
  }
#pragma unroll
  for (int n = 0; n < NST; ++n) {
    const size_t idx = (size_t)(((s * BB + b) * NST + n)) * D2C + d;
    Pbuf[idx] = P[n];
    qbuf[idx] = q[n];
  }
}

// Pass 2: stitch segments (sequential over SEG=16 only).
__global__ __launch_bounds__(256) void scan_stitch_kernel(
    const float* __restrict__ Pbuf, const float* __restrict__ qbuf,
    float* __restrict__ hstart) {
  const int t = blockIdx.x * 256 + threadIdx.x;  // 4096
  const int d = t & 511;
  const int b = t >> 9;
  float h[NST];
#pragma unroll
  for (int n = 0; n < NST; ++n) h[n] = 0.0f;
  for (int s = 0; s < SEG; ++s) {
#pragma unroll
    for (int n = 0; n < NST; ++n) {
      const size_t idx = (size_t)(((s * BB + b) * NST + n)) * D2C + d;
      hstart[idx] = h[n];
      h[n] = Pbuf[idx] * h[n] + qbuf[idx];
    }
  }
}

// Pass 3: recompute local scans from stitched initial states; emit y.
__global__ __launch_bounds__(256) void scan_emit_kernel(
    const float* __restrict__ delta, const float* __restrict__ u,
    const float* __restrict__ xdbl, const float* __restrict__ A_log,
    const float* __restrict__ Dvec, const float* __restrict__ hstart,
    float* __restrict__ yz) {
  const int t = blockIdx.x * 256 + threadIdx.x;  // 65536
  const int d = t & 511;
  const int b = (t >> 9) & 7;
  const int s = t >> 12;
  float Areg[NST], h[NST];
#pragma unroll
  for (int n = 0; n < NST; ++n) {
    Areg[n] = -expf(A_log[d * NST + n]);
    h[n] = hstart[(size_t)(((s * BB + b) * NST + n)) * D2C + d];
  }
  const float Dp = Dvec[d];
  for (int l = s * SLEN; l < s * SLEN + SLEN; ++l) {
    const size_t row = (size_t)(b * LL + l);
    const float dl = delta[row * D2C + d];
    const float ul = u[row * D2C + d];
    const float du = dl * ul;
    const float* xd = xdbl + row * 64;
    float y = 0.0f;
#pragma unroll
    for (int n = 0; n < NST; ++n) {
      const float dA = expf(dl * Areg[n]);
      h[n] = dA * h[n] + du * xd[DTR + n];
      y += h[n] * xd[DTR + NST + n];
    }
    yz[row * DI + d] = y + Dp * ul;
  }
}

// ---------------- launcher ----------------
extern "C" void kernel_launch(void* const* d_in, const int* in_sizes, int n_in,
                              void* d_out, int out_size, void* d_ws, size_t ws_size,
                              hipStream_t stream) {
  const float* x         = (const float*)d_in[0];
  const float* ln1_w     = (const float*)d_in[1];
  const float* ln1_b     = (const float*)d_in[2];
  const float* in_proj_w = (const float*)d_in[3];   // (1024, 512)
  const float* conv_x_w  = (const float*)d_in[4];   // (512, 1, 4)
  const float* conv_z_w  = (const float*)d_in[5];
  const float* x_proj_w  = (const float*)d_in[6];   // (64, 512)
  const float* dt_proj_w = (const float*)d_in[7];   // (512, 32)
  const float* dt_proj_b = (const float*)d_in[8];   // (512,)
  const float* A_log     = (const float*)d_in[9];   // (512, 16)
  const float* ssm_D     = (const float*)d_in[10];  // (512,)
  const float* out_proj_w= (const float*)d_in[11];  // (512, 1024)
  const float* ln2_w     = (const float*)d_in[12];
  const float* ln2_b     = (const float*)d_in[13];
  const float* fc1_w     = (const float*)d_in[14];  // (2048, 512)
  const float* fc1_b     = (const float*)d_in[15];
  const float* fc2_w     = (const float*)d_in[16];  // (512, 2048)
  const float* fc2_b     = (const float*)d_in[17];
  float* out = (float*)d_out;

  // workspace layout (float offsets), with reuse; total ~142 MB
  float* ws     = (float*)d_ws;
  float* xn     = ws;                         // (M,512)  [reused as h2]
  float* xz     = ws + (size_t)4194304;       // (M,1024) [reused as mlp1]
  float* xh     = ws + (size_t)12582912;      // (M,512)
  float* yz     = ws + (size_t)16777216;      // (M,1024) [y | zh]
  float* xdbl   = ws + (size_t)25165824;      // (M,64)
  float* delta  = ws + (size_t)25690112;      // (M,512)
  float* hbuf   = ws + (size_t)29884416;      // (M,512)
  float* Pbuf   = ws + (size_t)34078720;      // (16,8,16,512)
  float* qbuf   = ws + (size_t)35127296;      // (16,8,16,512)
  float* hstart = ws + (size_t)36175872;      // (16,8,16,512)
  float* h2     = xn;
  float* mlp1   = xz;

  const dim3 blk(256);

  // 1) LN1
  ln_kernel<<<MROWS, 128, 0, stream>>>(x, ln1_w, ln1_b, xn);
  // 2) in_proj: xz = xn @ in_proj_w^T            M=8192,K=512,N=1024
  gemm_wmma_f32<0, 64><<<dim3(1024 / 64, MROWS / 128), blk, 0, stream>>>(
      xn, DM, in_proj_w, DM, DM, nullptr, nullptr, xz, DI);
  // 3) depthwise conv + SiLU -> xh, zh(into yz right half)
  conv_silu_kernel<<<(MROWS * D2C) / 256, blk, 0, stream>>>(
      xz, conv_x_w, conv_z_w, xh, yz);
  // 4) x_proj: xdbl = xh @ x_proj_w^T            N=64
  gemm_wmma_f32<0, 64><<<dim3(64 / 64, MROWS / 128), blk, 0, stream>>>(
      xh, D2C, x_proj_w, D2C, D2C, nullptr, nullptr, xdbl, 64);
  // 5) dt_proj + bias + softplus: delta          K=32 (lda=64), N=512
  gemm_wmma_f32<1, 32><<<dim3(512 / 64, MROWS / 128), blk, 0, stream>>>(
      xdbl, 64, dt_proj_w, DTR, DTR, dt_proj_b, nullptr, delta, D2C);
  // 6) segmented selective scan -> y into yz left half
  scan_seg_kernel<<<(SEG * BB * D2C) / 256, blk, 0, stream>>>(
      delta, xh, xdbl, A_log, Pbuf, qbuf);
  scan_stitch_kernel<<<(BB * D2C) / 256, blk, 0, stream>>>(Pbuf, qbuf, hstart);
  scan_emit_kernel<<<(SEG * BB * D2C) / 256, blk, 0, stream>>>(
      delta, xh, xdbl, A_log, ssm_D, hstart, yz);
  // 7) out_proj + residual x: hbuf               K=1024, N=512
  gemm_wmma_f32<2, 64><<<dim3(512 / 64, MROWS / 128), blk, 0, stream>>>(
      yz, DI, out_proj_w, DI, DI, nullptr, x, hbuf, DM);
  // 8) LN2
  ln_kernel<<<MROWS, 128, 0, stream>>>(hbuf, ln2_w, ln2_b, h2);
  // 9) fc1 + bias + GELU: mlp1                   N=2048
  gemm_wmma_f32<3, 64><<<dim3(HID / 64, MROWS / 128), blk, 0, stream>>>(
      h2, DM, fc1_w, DM, DM, fc1_b, nullptr, mlp1, HID);
  // 10) fc2 + bias + residual h: out             K=2048, N=512
  gemm_wmma_f32<4, 64><<<dim3(512 / 64, MROWS / 128), blk, 0, stream>>>(
      mlp1, HID, fc2_w, HID, HID, fc2_b, hbuf, out, DM);
}